// SKA1d_46042049413166
// MI455X (gfx1250) — compile-verified
//
#include <hip/hip_runtime.h>
#include <stdint.h>

// Problem constants (from the reference)
#define BSZ   8
#define CCH   256
#define LLEN  8192
#define GGRP  64
#define KTAP  7
#define PAD   3
#define CPG   4              // channels per group = C/G
#define TILE  1024           // l-columns per block
#define XS    (TILE + 8)     // x row stride in LDS: 3+3 halo + 2 slack, 16B-aligned

typedef unsigned int       u32;
typedef unsigned long long u64;
typedef u32 u32x4 __attribute__((ext_vector_type(4)));
typedef int i32x4 __attribute__((ext_vector_type(4)));
typedef int i32x8 __attribute__((ext_vector_type(8)));

// Issue one TDM tensor_load_to_lds: 2D tile (tile_d1 rows x tile_d0 cols of 4B
// elements), row stride s0_elems, exact dims (no OOB clipping needed).
// D# layout per CDNA5 ISA ch.8 (group0 128b, group1 256b; groups 2/3 zero).
// This toolchain's builtin is the 6-arg form:
//   (uint32x4 g0, int32x8 g1, int32x4, int32x4, int32x8, i32 cpol)
static __device__ __forceinline__ void tdm_load_tile(u32 lds_addr, const void* gptr,
                                                     u32 tile_d0, u32 tile_d1,
                                                     u64 s0_elems)
{
    const u64 ga = (u64)(uintptr_t)gptr;

    u32x4 g0;
    g0.x = 1u;                                           // count=1, user descriptor
    g0.y = lds_addr;                                     // LDS byte address
    g0.z = (u32)ga;                                      // global_addr[31:0]
    g0.w = (u32)((ga >> 32) & 0x1FFFFFFu) | (2u << 30);  // global_addr[56:32] | type=2

    const u32 td0 = tile_d0;     // tensor_dim0 == tile_dim0 (exact)
    const u32 td1 = tile_d1;     // tensor_dim1 == tile_dim1 (exact)

    i32x8 g1;
    g1[0] = (int)(2u << 16);                                  // data_size=2 (4B), mask=0
    g1[1] = (int)((td0 & 0xFFFFu) << 16);                     // abar_addr=0 | tdim0.lo
    g1[2] = (int)((td0 >> 16) | ((td1 & 0xFFFFu) << 16));     // tdim0.hi | tdim1.lo
    g1[3] = (int)((td1 >> 16) | (tile_d0 << 16));             // tdim1.hi | tile_dim0
    g1[4] = (int)(tile_d1 & 0xFFFFu);                         // tile_dim1 | tile_dim2=0
    g1[5] = (int)(u32)s0_elems;                               // dim0_stride[31:0]
    g1[6] = (int)((u32)(s0_elems >> 32) & 0xFFFFu);           // dim0_stride[47:32] | d1s.lo=0
    g1[7] = 0;                                                // dim1_stride.hi = 0

    const i32x4 z4 = {0, 0, 0, 0};
    const i32x8 z8 = {0, 0, 0, 0, 0, 0, 0, 0};
    __builtin_amdgcn_tensor_load_to_lds(g0, g1, z4, z4, z8, 0);
}

__global__ __launch_bounds__(256) void ska1d_tdm_kernel(const float* __restrict__ x,
                                                        const float* __restrict__ w,
                                                        float* __restrict__ out)
{
    __shared__ __align__(16) float ws_s[KTAP * TILE];   // 28 KB: w taps for this (b,g) tile
    __shared__ __align__(16) float xs_s[CPG * XS];      // 16.1 KB: 4 channels + halo

    const int tid  = (int)threadIdx.x;
    const int tile = (int)blockIdx.x;
    const int g    = (int)blockIdx.y;
    const int b    = (int)blockIdx.z;
    const int l0   = tile * TILE;

    const int lo  = (l0 - PAD) > 0 ? (l0 - PAD) : 0;
    const int hi  = (l0 + TILE + PAD) < LLEN ? (l0 + TILE + PAD) : LLEN;
    const int off = lo - (l0 - PAD);   // 0..3 (left-edge clamp)
    const int cnt = hi - lo;           // columns the DMA actually fetches

    // Zero the LDS halo slots the DMA does not cover (disjoint from DMA region).
    if (tid < 64) {
        const int j = tid >> 4;
        const int i = tid & 15;
        if (i < off) xs_s[j * XS + i] = 0.0f;
        const int t = XS - 16 + i;
        if (t >= off + cnt) xs_s[j * XS + t] = 0.0f;
    }

    // Wave 0 (scalar-uniform branch) drives the Tensor Data Mover.
    if (__builtin_amdgcn_readfirstlane((int)threadIdx.x) == 0) {
        const u32 ws_base = (u32)(uintptr_t)&ws_s[0];
        const u32 xs_base = (u32)(uintptr_t)&xs_s[0];

        // w tile: 7 rows x TILE cols, row stride L elements; rows packed in LDS.
        const float* wsrc = w + ((u64)((b * GGRP + g) * KTAP)) * LLEN + l0;
        tdm_load_tile(ws_base, wsrc, TILE, KTAP, (u64)LLEN);

        // x: one 1D row per channel of the group (c = g + 64*j).
        #pragma unroll
        for (int j = 0; j < CPG; ++j) {
            const float* xsrc = x + ((u64)(b * CCH + g + GGRP * j)) * LLEN + lo;
            const u32 lds = xs_base + (u32)(j * XS + off) * 4u;
            tdm_load_tile(lds, xsrc, (u32)cnt, 1u, 0ull);
        }
        __builtin_amdgcn_s_wait_tensorcnt(0);
    }
    __syncthreads();   // publish TDM-written LDS to all 8 waves

    // Each thread: float4 of outputs at columns p..p+3, for all 4 channels.
    const int p = tid * 4;

    float wv[KTAP][4];
    #pragma unroll
    for (int k = 0; k < KTAP; ++k) {
        const float4 t = *(const float4*)&ws_s[k * TILE + p];
        wv[k][0] = t.x; wv[k][1] = t.y; wv[k][2] = t.z; wv[k][3] = t.w;
    }

    const u64 obase = ((u64)(b * CCH + g)) * LLEN + (u64)(l0 + p);

    #pragma unroll
    for (int j = 0; j < CPG; ++j) {
        const float* xr = &xs_s[j * XS + p];
        const float4 a = *(const float4*)(xr);       // xs[p..p+3]   (16B aligned)
        const float4 c = *(const float4*)(xr + 4);   // xs[p+4..p+7] (16B aligned)
        const float2 e = *(const float2*)(xr + 8);   // xs[p+8..p+9] (8B aligned)
        const float win[10] = {a.x, a.y, a.z, a.w, c.x, c.y, c.z, c.w, e.x, e.y};

        float4 acc;
        acc.x = 0.0f; acc.y = 0.0f; acc.z = 0.0f; acc.w = 0.0f;
        #pragma unroll
        for (int k = 0; k < KTAP; ++k) {
            acc.x = fmaf(win[k + 0], wv[k][0], acc.x);
            acc.y = fmaf(win[k + 1], wv[k][1], acc.y);
            acc.z = fmaf(win[k + 2], wv[k][2], acc.z);
            acc.w = fmaf(win[k + 3], wv[k][3], acc.w);
        }
        *(float4*)(out + obase + (u64)j * GGRP * LLEN) = acc;
    }
}

extern "C" void kernel_launch(void* const* d_in, const int* in_sizes, int n_in,
                              void* d_out, int out_size, void* d_ws, size_t ws_size,
                              hipStream_t stream)
{
    const float* x = (const float*)d_in[0];   // (B, C, L) f32
    const float* w = (const float*)d_in[1];   // (B, G, K, L) f32
    float* out     = (float*)d_out;           // (B, C, L) f32

    dim3 grid(LLEN / TILE, GGRP, BSZ);        // (8, 64, 8) = 4096 blocks
    ska1d_tdm_kernel<<<grid, 256, 0, stream>>>(x, w, out);
}